// Order_Constraint_video_6923487282636
// MI455X (gfx1250) — compile-verified
//
#include <hip/hip_runtime.h>

typedef float v2f __attribute__((ext_vector_type(2)));
typedef float v8f __attribute__((ext_vector_type(8)));

#define G 16
#define TILE 16
#define CCH 128
#define EPSF 1e-12f

// ---------------- workspace layout (floats) ----------------
// [0 .. 2047]   gsum[16][128]
// [2048..2063]  counts[16]
// [2064..2079]  sqsum[16]
#define WS_FLOATS (G*CCH + 2*G)

__global__ __launch_bounds__(256)
void oc_init_ws(float* __restrict__ ws) {
    int i = blockIdx.x * 256 + threadIdx.x;
    if (i < WS_FLOATS) ws[i] = 0.0f;
}

// ------------- main streaming pass: one-hot WMMA segment sum -------------
__global__ __launch_bounds__(256)
void oc_partial(const float* __restrict__ feat, const int* __restrict__ scores,
                float* __restrict__ gsum, float* __restrict__ counts,
                float* __restrict__ sqsum, int B) {
    __shared__ float ldsF[TILE * CCH];   // 16 rows x 128 chans (8 KB)
    __shared__ float ldsA[TILE * G];     // A matrix [k][g] = scaled one-hot (1 KB)
    __shared__ float ldsS[TILE];         // row sum-of-squares
    __shared__ float ldsCnt[G];
    __shared__ float ldsSq[G];

    const int t      = threadIdx.x;
    const int wave   = t >> 5;           // 0..7 -> channel slice
    const int lane   = t & 31;
    const int laneLo = lane & 15;
    const int hiHalf = lane >> 4;        // 0 or 1 (K half within wmma)
    const int c0     = wave * 16;        // this wave's 16-channel slice

    if (t < G) { ldsCnt[t] = 0.0f; ldsSq[t] = 0.0f; }

    v8f acc = {0.f, 0.f, 0.f, 0.f, 0.f, 0.f, 0.f, 0.f};

    const int numTiles = (B + TILE - 1) / TILE;
    for (int tile = blockIdx.x; tile < numTiles; tile += gridDim.x) {
        const int rowBase = tile * TILE;
        __syncthreads();   // protect LDS reuse across iterations

        // ---- stage 16 rows into LDS (fully coalesced: 2 float4 per thread) ----
        float4 v0, v1;
        {
            const int f4a  = 2 * t;            // float4 index within tile
            const int row  = f4a >> 5;         // = t/16
            const int gRow = rowBase + row;
            if (gRow < B) {
                const float4* gp = (const float4*)(feat + (size_t)gRow * CCH);
                const int col4 = f4a & 31;
                v0 = gp[col4];
                v1 = gp[col4 + 1];
            } else {
                v0 = make_float4(0.f, 0.f, 0.f, 0.f);
                v1 = v0;
            }
            float4* lp = (float4*)ldsF;
            lp[f4a]     = v0;
            lp[f4a + 1] = v1;
        }
        // ---- row sum-of-squares: reduce within each 16-lane half (one row each) ----
        float sq = v0.x*v0.x + v0.y*v0.y + v0.z*v0.z + v0.w*v0.w
                 + v1.x*v1.x + v1.y*v1.y + v1.z*v1.z + v1.w*v1.w;
        sq += __shfl_xor(sq, 1);
        sq += __shfl_xor(sq, 2);
        sq += __shfl_xor(sq, 4);
        sq += __shfl_xor(sq, 8);
        if (laneLo == 0) ldsS[2 * wave + hiHalf] = sq;
        __syncthreads();

        // ---- build A[k][g] = (score==g) ? 1/max(||x||,eps) : 0 ; counts/sqsum ----
        {
            const int k = t >> 4, g = t & 15;
            const float s   = ldsS[k];
            const float n   = sqrtf(s);
            const float inv = 1.0f / fmaxf(n, EPSF);
            const int   gr  = rowBase + k;
            const int   sc  = (gr < B) ? scores[gr] : -1;
            ldsA[k * G + g] = (sc == g) ? inv : 0.0f;
            if (g == 0 && sc >= 0) {
                atomicAdd(&ldsCnt[sc], 1.0f);            // ds_add_f32
                atomicAdd(&ldsSq[sc], s * inv * inv);    // ||f||^2 contribution
            }
        }
        __syncthreads();

        // ---- 4x V_WMMA_F32_16X16X4_F32: D[g][n] += A[g][k] * X[k][n] ----
        #pragma unroll
        for (int s4 = 0; s4 < 4; ++s4) {
            const int k0 = 4 * s4 + 2 * hiHalf;   // K half-split per ISA layout
            v2f a, b;
            a.x = ldsA[k0 * G + laneLo];
            a.y = ldsA[(k0 + 1) * G + laneLo];
            b.x = ldsF[k0 * CCH + c0 + laneLo];
            b.y = ldsF[(k0 + 1) * CCH + c0 + laneLo];
            acc = __builtin_amdgcn_wmma_f32_16x16x4_f32(
                      false, a, false, b, (short)0, acc, false, false);
        }
    }

    // ---- flush per-wave group-sum tile (C/D layout: M=r / r+8 per half-wave) ----
    {
        const int n     = c0 + laneLo;
        const int mBase = hiHalf * 8;
        #pragma unroll
        for (int r = 0; r < 8; ++r)
            atomicAdd(&gsum[(mBase + r) * CCH + n], acc[r]);
    }
    __syncthreads();
    if (t < G)            atomicAdd(&counts[t], ldsCnt[t]);
    else if (t < 2 * G)   atomicAdd(&sqsum[t - G], ldsSq[t - G]);
}

// ------------- finalize: compact present groups, 14-term loss -------------
__global__ __launch_bounds__(128)
void oc_final(const float* __restrict__ gsum, const float* __restrict__ counts,
              const float* __restrict__ sqsum, float* __restrict__ out) {
    __shared__ float red[CCH];
    const int t = threadIdx.x;

    // remap: list of present original groups, in order (computed redundantly)
    int idxs[G];
    int U = 0;
    #pragma unroll
    for (int g = 0; g < G; ++g)
        if (counts[g] > 0.0f) idxs[U++] = g;

    // per-channel partial of sum_i [ m2*(m1+m3) - m1*m3 ]
    float chanTerm = 0.0f;
    for (int i = 0; i < G - 2; ++i) {
        if (i + 2 < U) {
            const int g1 = idxs[i], g2 = idxs[i + 1], g3 = idxs[i + 2];
            const float m1 = gsum[g1 * CCH + t] / fmaxf(counts[g1], 1.0f);
            const float m2 = gsum[g2 * CCH + t] / fmaxf(counts[g2], 1.0f);
            const float m3 = gsum[g3 * CCH + t] / fmaxf(counts[g3], 1.0f);
            chanTerm += m2 * (m1 + m3) - m1 * m3;
        }
    }
    red[t] = chanTerm;
    __syncthreads();
    for (int off = 64; off > 0; off >>= 1) {
        if (t < off) red[t] += red[t + off];
        __syncthreads();
    }
    if (t == 0) {
        float msq = 0.0f;
        for (int i = 0; i < G - 2; ++i) {
            if (i + 2 < U) {
                const int g2 = idxs[i + 1];
                msq += sqsum[g2] / fmaxf(counts[g2], 1.0f);
            }
        }
        out[0] = (red[0] - msq) / (float)(U - 2);
    }
}

extern "C" void kernel_launch(void* const* d_in, const int* in_sizes, int n_in,
                              void* d_out, int out_size, void* d_ws, size_t ws_size,
                              hipStream_t stream) {
    const float* feat   = (const float*)d_in[0];
    const int*   scores = (const int*)d_in[1];
    const int    B      = in_sizes[1];

    float* ws     = (float*)d_ws;
    float* gsum   = ws;                // 2048 floats
    float* counts = ws + G * CCH;      // 16
    float* sqsum  = counts + G;        // 16

    oc_init_ws<<<(WS_FLOATS + 255) / 256, 256, 0, stream>>>(ws);

    const int numTiles = (B + TILE - 1) / TILE;
    int grid = 1024;
    if (grid > numTiles) grid = numTiles;
    oc_partial<<<grid, 256, 0, stream>>>(feat, scores, gsum, counts, sqsum, B);

    oc_final<<<1, CCH, 0, stream>>>(gsum, counts, sqsum, (float*)d_out);
}